// NKRR_10393820856368
// MI455X (gfx1250) — compile-verified
//
#include <hip/hip_runtime.h>

// CDNA5 / gfx1250, wave32. Uses V_WMMA_F32_16X16X4_F32 (exact fp32 matrix math).
typedef float v2f __attribute__((ext_vector_type(2)));
typedef float v8f __attribute__((ext_vector_type(8)));

__device__ __forceinline__ float wshfl(float v, int src) { return __shfl(v, src, 32); }
__device__ __forceinline__ float wshflx(float v, int m)  { return __shfl_xor(v, m, 32); }

// Load one 16-column B tile (centers rows [jbase, jbase+16)) into per-lane
// WMMA f32 16x16x4 operand layout, plus the column norm and alpha value.
//   B (4x16): lane = 16*half + col ; VGPR v holds K = 2*half + v
__device__ __forceinline__ void load_btile(const float* __restrict__ C,
                                           const float* __restrict__ alpha,
                                           int c, int half,
                                           v2f b[16], float& c2, float& al) {
    const float* bp = C + (size_t)c * 64 + 2 * half;
    float cs = 0.0f;
#pragma unroll
    for (int t = 0; t < 16; ++t) {
        float bx = bp[4 * t + 0];
        float by = bp[4 * t + 1];
        b[t].x = bx; b[t].y = by;
        cs += bx * bx + by * by;
    }
    c2 = cs + wshflx(cs, 16);   // full ||C_c||^2, valid in all lanes
    al = alpha[c];
}

// One wave computes, for its 16 rows [rbase, rbase+16) of A (clamped to nA),
//   pred[p] = sum_j exp(-0.5*||A_i - C_j||^2 / sigma^2) * alpha[j]
// for i = rbase + 8*half + p, p = 0..7 (result replicated across the lane's
// 16-lane half after the final shuffle reduction). D fixed at 64; M % 16 == 0.
//
// WMMA f32 16x16x4 operand layout (ISA 7.12.2):
//   A (16x4): lane = 16*half + row ; VGPR v holds K = 2*half + v
//   C/D     : VGPR p, lanes 0-15 -> M=p,  lanes 16-31 -> M=8+p ; N = lane&15
//
// Two j-tiles are processed per main-loop iteration with independent
// accumulators so the v_wmma D->C chains interleave instead of serializing.
__device__ void krr_wave_rows(const float* __restrict__ A, int nA, int rbase,
                              const float* __restrict__ C, int M,
                              const float* __restrict__ alpha,
                              float inv_s2, float pred[8]) {
    const int lane = threadIdx.x & 31;
    const int half = lane >> 4;   // 0 or 1
    const int lcol = lane & 15;   // row (for A loads) / col (for B loads, C/D N)
    const float nh = -0.5f * inv_s2;

    // ---- Preload this wave's 16 A rows (each lane: 32 of the 64 values) ----
    int arow = rbase + lcol;
    if (arow > nA - 1) arow = nA - 1;                 // clamp; masked later
    const float* ap = A + (size_t)arow * 64 + 2 * half;
    v2f a[16];
    float s = 0.0f;
#pragma unroll
    for (int t = 0; t < 16; ++t) {
        float ax = ap[4 * t + 0];
        float ay = ap[4 * t + 1];
        a[t].x = ax; a[t].y = ay;
        s += ax * ax + ay * ay;
    }
    float x2row = s + wshflx(s, 16);                  // x2[rbase + lcol], all lanes
    float x2p[8];
#pragma unroll
    for (int p = 0; p < 8; ++p) x2p[p] = wshfl(x2row, 8 * half + p);

#pragma unroll
    for (int p = 0; p < 8; ++p) pred[p] = 0.0f;

    // ---- Main loop: two 16-column center tiles per iteration ----
    int j0 = 0;
    for (; j0 + 32 <= M; j0 += 32) {
        v2f b0[16], b1[16];
        float c20, c21, al0, al1;
        load_btile(C, alpha, j0 + lcol,      half, b0, c20, al0);
        load_btile(C, alpha, j0 + 16 + lcol, half, b1, c21, al1);

        v8f acc0 = {0.0f, 0.0f, 0.0f, 0.0f, 0.0f, 0.0f, 0.0f, 0.0f};
        v8f acc1 = {0.0f, 0.0f, 0.0f, 0.0f, 0.0f, 0.0f, 0.0f, 0.0f};
#pragma unroll
        for (int t = 0; t < 16; ++t) {
            acc0 = __builtin_amdgcn_wmma_f32_16x16x4_f32(
                false, a[t], false, b0[t], (short)0, acc0, false, false);
            acc1 = __builtin_amdgcn_wmma_f32_16x16x4_f32(
                false, a[t], false, b1[t], (short)0, acc1, false, false);
        }
#pragma unroll
        for (int p = 0; p < 8; ++p) {
            float sq0 = x2p[p] + c20 - 2.0f * acc0[p];
            float sq1 = x2p[p] + c21 - 2.0f * acc1[p];
            float w0 = __expf(sq0 * nh);
            float w1 = __expf(sq1 * nh);
            pred[p] += w0 * al0 + w1 * al1;
        }
    }
    // ---- Tail: at most one 16-column tile ----
    for (; j0 < M; j0 += 16) {
        v2f b0[16];
        float c20, al0;
        load_btile(C, alpha, j0 + lcol, half, b0, c20, al0);
        v8f acc0 = {0.0f, 0.0f, 0.0f, 0.0f, 0.0f, 0.0f, 0.0f, 0.0f};
#pragma unroll
        for (int t = 0; t < 16; ++t) {
            acc0 = __builtin_amdgcn_wmma_f32_16x16x4_f32(
                false, a[t], false, b0[t], (short)0, acc0, false, false);
        }
#pragma unroll
        for (int p = 0; p < 8; ++p) {
            float sq0 = x2p[p] + c20 - 2.0f * acc0[p];
            pred[p] += __expf(sq0 * nh) * al0;
        }
    }

    // ---- Row sums: reduce across the 16 columns within each half ----
#pragma unroll
    for (int p = 0; p < 8; ++p) {
        float v = pred[p];
        v += wshflx(v, 8);
        v += wshflx(v, 4);
        v += wshflx(v, 2);
        v += wshflx(v, 1);
        pred[p] = v;
    }
}

// ---------------- Kernel 1: preds = K_NM @ alpha, + per-block SSE ----------------
__global__ __launch_bounds__(256) void nkrr_preds_kernel(
    const float* __restrict__ X, const float* __restrict__ Y,
    const float* __restrict__ sigma,
    const float* __restrict__ centers, const float* __restrict__ alpha,
    float* __restrict__ preds_out, float* __restrict__ loss_partial,
    int N, int M) {
    const int wave  = threadIdx.x >> 5;
    const int rbase = blockIdx.x * 128 + wave * 16;
    const float sg = sigma[0];
    const float inv_s2 = 1.0f / (sg * sg);

    float pred[8];
    krr_wave_rows(X, N, rbase, centers, M, alpha, inv_s2, pred);

    const int lane = threadIdx.x & 31;
    const int half = lane >> 4;
    const int lcol = lane & 15;
    float lsum = 0.0f;
    if (lcol == 0) {
#pragma unroll
        for (int p = 0; p < 8; ++p) {
            int i = rbase + 8 * half + p;
            if (i < N) {
                float pv = pred[p];
                preds_out[i] = pv;
                float d = pv - Y[i];
                lsum += d * d;
            }
        }
    }
    // deterministic block reduction (8 waves)
    __shared__ float red[8];
    for (int m = 16; m >= 1; m >>= 1) lsum += wshflx(lsum, m);
    if (lane == 0) red[wave] = lsum;
    __syncthreads();
    if (threadIdx.x == 0) {
        float t = 0.0f;
        for (int w = 0; w < 8; ++w) t += red[w];
        loss_partial[blockIdx.x] = t;
    }
}

// ---------------- Kernel 2: per-block partials of alpha^T K_MM alpha ----------------
__global__ __launch_bounds__(256) void nkrr_reg_kernel(
    const float* __restrict__ centers, const float* __restrict__ sigma,
    const float* __restrict__ alpha, float* __restrict__ reg_partial, int M) {
    const int wave  = threadIdx.x >> 5;
    const int rbase = blockIdx.x * 128 + wave * 16;
    const float sg = sigma[0];
    const float inv_s2 = 1.0f / (sg * sg);

    float pred[8];   // pred[p] = (K_MM @ alpha)[row]
    krr_wave_rows(centers, M, rbase, centers, M, alpha, inv_s2, pred);

    const int lane = threadIdx.x & 31;
    const int half = lane >> 4;
    const int lcol = lane & 15;
    float rsum = 0.0f;
    if (lcol == 0) {
#pragma unroll
        for (int p = 0; p < 8; ++p) {
            int i = rbase + 8 * half + p;
            if (i < M) rsum += alpha[i] * pred[p];
        }
    }
    __shared__ float red[8];
    for (int m = 16; m >= 1; m >>= 1) rsum += wshflx(rsum, m);
    if (lane == 0) red[wave] = rsum;
    __syncthreads();
    if (threadIdx.x == 0) {
        float t = 0.0f;
        for (int w = 0; w < 8; ++w) t += red[w];
        reg_partial[blockIdx.x] = t;
    }
}

// ---------------- Kernel 3: sequential, deterministic finalize ----------------
__global__ void nkrr_finalize(const float* __restrict__ loss_partial, int nb1,
                              const float* __restrict__ reg_partial, int nb2,
                              const float* __restrict__ penalty,
                              float* __restrict__ out0, float invN) {
    if (blockIdx.x == 0 && threadIdx.x == 0) {
        float l = 0.0f;
        for (int i = 0; i < nb1; ++i) l += loss_partial[i];
        float r = 0.0f;
        for (int i = 0; i < nb2; ++i) r += reg_partial[i];
        out0[0] = l * invN + __expf(-penalty[0]) * r;
    }
}

extern "C" void kernel_launch(void* const* d_in, const int* in_sizes, int n_in,
                              void* d_out, int out_size, void* d_ws, size_t ws_size,
                              hipStream_t stream) {
    // setup_inputs order: X, Y, sigma, penalty, centers, alpha
    const float* X       = (const float*)d_in[0];
    const float* Y       = (const float*)d_in[1];
    const float* sigma   = (const float*)d_in[2];
    const float* penalty = (const float*)d_in[3];
    const float* centers = (const float*)d_in[4];
    const float* alpha   = (const float*)d_in[5];

    const int D = 64;
    const int N = in_sizes[0] / D;   // 100000
    const int M = in_sizes[4] / D;   // 2000 (multiple of 16)

    float* out   = (float*)d_out;    // out[0] = loss+reg ; out[1..N] = preds
    float* preds = out + 1;

    const int nb1 = (N + 127) / 128; // 782 blocks, 128 rows each
    const int nb2 = (M + 127) / 128; // 16 blocks

    float* lossp = (float*)d_ws;     // nb1 floats
    float* regp  = lossp + nb1;      // nb2 floats

    nkrr_preds_kernel<<<nb1, 256, 0, stream>>>(X, Y, sigma, centers, alpha,
                                               preds, lossp, N, M);
    nkrr_reg_kernel<<<nb2, 256, 0, stream>>>(centers, sigma, alpha, regp, M);
    nkrr_finalize<<<1, 1, 0, stream>>>(lossp, nb1, regp, nb2, penalty,
                                       out, 1.0f / (float)N);
}